// GCNEncoder_37434934952020
// MI455X (gfx1250) — compile-verified
//
#include <hip/hip_runtime.h>
#include <float.h>
#include <math.h>

typedef __attribute__((ext_vector_type(2))) float v2f;
typedef __attribute__((ext_vector_type(8))) float v8f;

#define NEG_SLOPE 0.2f
#define GAT_EPS 1e-16f

// ---------------------------------------------------------------- utilities

__device__ __forceinline__ void atomicMaxF(float* addr, float val) {
  // Order-preserving float max via signed-int max (for >=0) / unsigned-int min
  // (for <0). Safe for mixed signs; init value must be -FLT_MAX.
  if (val >= 0.0f) atomicMax((int*)addr, __float_as_int(val));
  else             atomicMin((unsigned int*)addr, __float_as_uint(val));
}

__global__ void fill_f32(float* __restrict__ p, float v, size_t n) {
  size_t t = (size_t)blockIdx.x * blockDim.x + threadIdx.x;
  if (t < n) p[t] = v;
}

// ------------------------------------------------- fp32 WMMA GEMM (16x16x4)
// D[M,Nc] = A[M,K] @ B[K,Nc]; row-major; one wave per 16x16 output tile.
// Requires K%4==0, Nc%16==0; waves past ceil(M/16) exit wave-uniformly so
// EXEC is all-ones for every executed v_wmma (ISA requirement).
__global__ void wmma_gemm_f32(const float* __restrict__ A,
                              const float* __restrict__ B,
                              float* __restrict__ D,
                              int M, int K, int Nc) {
  int wave   = (int)((blockIdx.x * (size_t)blockDim.x + threadIdx.x) >> 5);
  int lane   = threadIdx.x & 31;
  int tilesN = Nc >> 4;
  int mT     = wave / tilesN;
  int nT     = wave - mT * tilesN;
  if (mT * 16 >= M) return;                       // wave-uniform exit

  int half = lane >> 4;                           // 0: K pair {0,1}; 1: {2,3}
  int mr   = lane & 15;

  const float* Arow = A + (size_t)(mT * 16 + mr) * K + 2 * half;
  const float* Bcol = B + (size_t)nT * 16 + mr;

  v8f acc = {0.f, 0.f, 0.f, 0.f, 0.f, 0.f, 0.f, 0.f};

  #pragma unroll 4
  for (int k0 = 0; k0 < K; k0 += 4) {
    v2f a, b;
    a.x = Arow[k0];                               // A[m][k0 + 2*half]
    a.y = Arow[k0 + 1];                           // A[m][k0 + 2*half + 1]
    size_t kb = (size_t)(k0 + 2 * half) * Nc;
    b.x = Bcol[kb];                               // B[k0+2*half    ][n]
    b.y = Bcol[kb + Nc];                          // B[k0+2*half + 1][n]
    acc = __builtin_amdgcn_wmma_f32_16x16x4_f32(
        /*neg_a=*/false, a, /*neg_b=*/false, b,
        /*c_mod=*/(short)0, acc, /*reuse_a=*/false, /*reuse_b=*/false);
  }

  // acc[r] holds D[mT*16 + r + 8*half][nT*16 + mr]
  float* Dt = D + (size_t)(mT * 16 + 8 * half) * Nc + (size_t)nT * 16 + mr;
  #pragma unroll
  for (int r = 0; r < 8; ++r) Dt[(size_t)r * Nc] = acc[r];
}

// ------------------------------------------- per-node attention projections
// One block per node; warp w (of 8) handles head w; C channels per head.
__global__ void att_dots(const float* __restrict__ h,
                         const float* __restrict__ a_src,
                         const float* __restrict__ a_dst,
                         float* __restrict__ as, float* __restrict__ ad,
                         int C) {
  int n    = blockIdx.x;
  int warp = threadIdx.x >> 5;
  int lane = threadIdx.x & 31;
  const float* hr  = h + (size_t)n * 8 * C + (size_t)warp * C;
  const float* asr = a_src + (size_t)warp * C;
  const float* adr = a_dst + (size_t)warp * C;
  float ss = 0.f, sd = 0.f;
  for (int c = lane; c < C; c += 32) {
    float hv = hr[c];
    ss += hv * asr[c];
    sd += hv * adr[c];
  }
  #pragma unroll
  for (int off = 16; off > 0; off >>= 1) {
    ss += __shfl_down(ss, off);
    sd += __shfl_down(sd, off);
  }
  if (lane == 0) { as[(size_t)n * 8 + warp] = ss; ad[(size_t)n * 8 + warp] = sd; }
}

// ------------------------------------------------------------ edge kernels
__device__ __forceinline__ void edge_sd(const int* __restrict__ ei, int E,
                                        int i, int& s, int& d, float& mask) {
  if (i < E) { s = ei[i]; d = ei[E + i]; mask = (s != d) ? 1.0f : 0.0f; }
  else       { s = d = i - E;            mask = 1.0f; }          // self loop
}

__global__ void edge_max(const int* __restrict__ ei, int E, int N,
                         const float* __restrict__ as,
                         const float* __restrict__ ad,
                         float* __restrict__ m) {
  int i = (int)(blockIdx.x * (size_t)blockDim.x + threadIdx.x);
  if (i >= E + N) return;
  int s, d; float mask;
  edge_sd(ei, E, i, s, d, mask);
  #pragma unroll
  for (int h = 0; h < 8; ++h) {
    float e = as[(size_t)s * 8 + h] + ad[(size_t)d * 8 + h];
    e = (e > 0.f) ? e : NEG_SLOPE * e;            // leaky_relu
    atomicMaxF(&m[(size_t)d * 8 + h], e);         // max over ALL edges (pre-mask)
  }
}

__global__ void edge_exp(const int* __restrict__ ei, int E, int N,
                         const float* __restrict__ as,
                         const float* __restrict__ ad,
                         const float* __restrict__ m,
                         float* __restrict__ ex, float* __restrict__ denom) {
  int i = (int)(blockIdx.x * (size_t)blockDim.x + threadIdx.x);
  if (i >= E + N) return;
  int s, d; float mask;
  edge_sd(ei, E, i, s, d, mask);
  #pragma unroll
  for (int h = 0; h < 8; ++h) {
    float e = as[(size_t)s * 8 + h] + ad[(size_t)d * 8 + h];
    e = (e > 0.f) ? e : NEG_SLOPE * e;
    float v = __expf(e - m[(size_t)d * 8 + h]) * mask;
    ex[(size_t)i * 8 + h] = v;
    atomicAdd(&denom[(size_t)d * 8 + h], v);
  }
}

// ex[i,h] -> alpha[i,h] = ex[i,h] / (denom[dst(i),h] + eps)   (once per edge)
__global__ void normalize_alpha(const int* __restrict__ ei, int E, int N,
                                const float* __restrict__ denom,
                                float* __restrict__ ex) {
  int i = (int)(blockIdx.x * (size_t)blockDim.x + threadIdx.x);
  if (i >= E + N) return;
  int s, d; float mask;
  edge_sd(ei, E, i, s, d, mask);
  #pragma unroll
  for (int h = 0; h < 8; ++h)
    ex[(size_t)i * 8 + h] /= (denom[(size_t)d * 8 + h] + GAT_EPS);
}

// out[dst, j..j+3] += h[src, j..j+3] * alpha[i, head(j)]
// One thread per (edge, 4 channels): float4 (b128) gather, 4 atomic adds.
// F in {64, 1024}; channels/head = F/8 is a multiple of 4, so a group of 4
// never straddles a head boundary; rows are 16B-aligned.
__global__ void edge_agg4(const int* __restrict__ ei, int E, int N,
                          const float* __restrict__ hfeat,
                          const float* __restrict__ alpha,
                          float* __restrict__ out, int F) {
  size_t t = (size_t)blockIdx.x * blockDim.x + threadIdx.x;
  int Fq = F >> 2;
  size_t total = (size_t)(E + N) * (size_t)Fq;
  if (t >= total) return;
  int i = (int)(t / Fq);
  int q = (int)(t - (size_t)i * Fq);
  int j = q << 2;                                  // first of 4 channels
  int s, d; float mask;
  edge_sd(ei, E, i, s, d, mask);
  int h = j / (F >> 3);                            // head index
  float a = alpha[(size_t)i * 8 + h];
  const float4 hv = *(const float4*)(hfeat + (size_t)s * F + j);
  float* o = out + (size_t)d * F + j;
  atomicAdd(o + 0, hv.x * a);
  atomicAdd(o + 1, hv.y * a);
  atomicAdd(o + 2, hv.z * a);
  atomicAdd(o + 3, hv.w * a);
}

// ----------------------------------------------------------- pointwise ops
__global__ void bias_elu(const float* __restrict__ in,
                         const float* __restrict__ b,
                         float* __restrict__ out, size_t n, int F) {
  size_t t = (size_t)blockIdx.x * blockDim.x + threadIdx.x;
  if (t >= n) return;
  float v = in[t] + b[t % F];
  out[t] = (v > 0.f) ? v : (__expf(v) - 1.0f);     // ELU
}

// In-place: row = (row + b2) - logsumexp(row + b2); Nc == 1024; block = 256.
__global__ void finalize_logsoftmax(float* __restrict__ out,
                                    const float* __restrict__ b2) {
  const int Nc = 1024;
  int n = blockIdx.x;
  int t = threadIdx.x;
  float* row = out + (size_t)n * Nc;
  float v[4];
  float mx = -FLT_MAX;
  #pragma unroll
  for (int k = 0; k < 4; ++k) {
    v[k] = row[t + k * 256] + b2[t + k * 256];
    mx = fmaxf(mx, v[k]);
  }
  __shared__ float red[256];
  red[t] = mx; __syncthreads();
  for (int s = 128; s > 0; s >>= 1) {
    if (t < s) red[t] = fmaxf(red[t], red[t + s]);
    __syncthreads();
  }
  mx = red[0]; __syncthreads();
  float sum = 0.f;
  #pragma unroll
  for (int k = 0; k < 4; ++k) sum += __expf(v[k] - mx);
  red[t] = sum; __syncthreads();
  for (int s = 128; s > 0; s >>= 1) {
    if (t < s) red[t] += red[t + s];
    __syncthreads();
  }
  float lse = logf(red[0]) + mx;
  #pragma unroll
  for (int k = 0; k < 4; ++k) row[t + k * 256] = v[k] - lse;
}

// ---------------------------------------------------------------- launcher
static inline int cdiv_i(size_t a, size_t b) { return (int)((a + b - 1) / b); }

extern "C" void kernel_launch(void* const* d_in, const int* in_sizes, int n_in,
                              void* d_out, int out_size, void* d_ws, size_t ws_size,
                              hipStream_t stream) {
  const float* x    = (const float*)d_in[0];
  const int*   ei   = (const int*)d_in[1];      // [2,E] src row then dst row
  const float* W1   = (const float*)d_in[2];
  const float* a_s1 = (const float*)d_in[3];
  const float* a_d1 = (const float*)d_in[4];
  const float* b1   = (const float*)d_in[5];
  const float* W2   = (const float*)d_in[6];
  const float* a_s2 = (const float*)d_in[7];
  const float* a_d2 = (const float*)d_in[8];
  const float* b2   = (const float*)d_in[9];
  float* out = (float*)d_out;

  const int F  = 128;
  const int N  = in_sizes[0] / F;               // 10000
  const int E  = in_sizes[1] / 2;               // 160000
  const int Et = E + N;                         // with self loops
  const int H  = 8;
  const int F1 = 64;                            // H1*C1
  const int F2 = 1024;                          // H2*C2

  // workspace carve-up (floats)
  float* ws   = (float*)d_ws;
  float* h1   = ws; ws += (size_t)N * F1;
  float* as1  = ws; ws += (size_t)N * H;
  float* ad1  = ws; ws += (size_t)N * H;
  float* m1   = ws; ws += (size_t)N * H;
  float* den1 = ws; ws += (size_t)N * H;
  float* ex1  = ws; ws += (size_t)Et * H;
  float* agg1 = ws; ws += (size_t)N * F1;       // also holds ELU(h) in place
  float* h2   = ws; ws += (size_t)N * F2;
  float* as2  = ws; ws += (size_t)N * H;
  float* ad2  = ws; ws += (size_t)N * H;
  float* m2   = ws; ws += (size_t)N * H;
  float* den2 = ws; ws += (size_t)N * H;
  float* ex2  = ws; ws += (size_t)Et * H;

  const int B = 256;

  // ---- init accumulators
  fill_f32<<<cdiv_i((size_t)N * H, B), B, 0, stream>>>(m1, -FLT_MAX, (size_t)N * H);
  fill_f32<<<cdiv_i((size_t)N * H, B), B, 0, stream>>>(den1, 0.f, (size_t)N * H);
  fill_f32<<<cdiv_i((size_t)N * H, B), B, 0, stream>>>(m2, -FLT_MAX, (size_t)N * H);
  fill_f32<<<cdiv_i((size_t)N * H, B), B, 0, stream>>>(den2, 0.f, (size_t)N * H);
  fill_f32<<<cdiv_i((size_t)N * F1, B), B, 0, stream>>>(agg1, 0.f, (size_t)N * F1);
  fill_f32<<<cdiv_i((size_t)N * F2, B), B, 0, stream>>>(out, 0.f, (size_t)N * F2);

  // ---- layer 1: h1 = x @ W1   (fp32 WMMA)
  {
    size_t waves = (size_t)cdiv_i(N, 16) * (F1 / 16);
    wmma_gemm_f32<<<cdiv_i(waves * 32, B), B, 0, stream>>>(x, W1, h1, N, F, F1);
  }
  att_dots<<<N, B, 0, stream>>>(h1, a_s1, a_d1, as1, ad1, F1 / H);
  edge_max<<<cdiv_i(Et, B), B, 0, stream>>>(ei, E, N, as1, ad1, m1);
  edge_exp<<<cdiv_i(Et, B), B, 0, stream>>>(ei, E, N, as1, ad1, m1, ex1, den1);
  normalize_alpha<<<cdiv_i(Et, B), B, 0, stream>>>(ei, E, N, den1, ex1);
  edge_agg4<<<cdiv_i((size_t)Et * (F1 / 4), B), B, 0, stream>>>(ei, E, N, h1, ex1, agg1, F1);
  bias_elu<<<cdiv_i((size_t)N * F1, B), B, 0, stream>>>(agg1, b1, agg1, (size_t)N * F1, F1);

  // ---- layer 2: h2 = elu(h1agg) @ W2   (fp32 WMMA)
  {
    size_t waves = (size_t)cdiv_i(N, 16) * (F2 / 16);
    wmma_gemm_f32<<<cdiv_i(waves * 32, B), B, 0, stream>>>(agg1, W2, h2, N, F1, F2);
  }
  att_dots<<<N, B, 0, stream>>>(h2, a_s2, a_d2, as2, ad2, F2 / H);
  edge_max<<<cdiv_i(Et, B), B, 0, stream>>>(ei, E, N, as2, ad2, m2);
  edge_exp<<<cdiv_i(Et, B), B, 0, stream>>>(ei, E, N, as2, ad2, m2, ex2, den2);
  normalize_alpha<<<cdiv_i(Et, B), B, 0, stream>>>(ei, E, N, den2, ex2);
  edge_agg4<<<cdiv_i((size_t)Et * (F2 / 4), B), B, 0, stream>>>(ei, E, N, h2, ex2, out, F2);

  // ---- bias + log_softmax in place on d_out
  finalize_logsoftmax<<<N, B, 0, stream>>>(out, b2);
}